// LengthRegulator_31756988186732
// MI455X (gfx1250) — compile-verified
//
#include <hip/hip_runtime.h>

// ---------------------------------------------------------------------------
// LengthRegulator for MI455X (gfx1250).
//   B=32, TMAX=256, D=384, TOUT=2048, f32 in/out.
// Pure data movement: ~100.7 MB stores / call -> HBM-bound (~5-6 us floor at
// 23.3 TB/s). No matrix math => no WMMA; the CDNA5 feature that applies is the
// async global<->LDS copy path (ASYNCcnt) for the row gather.
// ---------------------------------------------------------------------------

#define LR_B     32
#define LR_TMAX  256
#define LR_D     384
#define LR_TOUT  2048

#define LR_ROWS_PER_BLOCK 8           // one output frame per wave32
#define LR_F4_PER_ROW     (LR_D / 4)  // 96 float4 per 1536B row
#define LR_F4_PER_LANE    (LR_F4_PER_ROW / 32) // 3

#define LR_AS1 __attribute__((address_space(1)))
#define LR_AS3 __attribute__((address_space(3)))

typedef int lr_v4i __attribute__((ext_vector_type(4)));

#if defined(__gfx1250__) &&                                                   \
    __has_builtin(__builtin_amdgcn_global_load_async_to_lds_b128) &&          \
    __has_builtin(__builtin_amdgcn_global_store_async_from_lds_b128) &&       \
    __has_builtin(__builtin_amdgcn_s_wait_asynccnt)
#define LR_ASYNC 1
#else
#define LR_ASYNC 0
#endif

__global__ __launch_bounds__(256) void
length_regulator_kernel(const float* __restrict__ emb,   // (B, TMAX, D)
                        const int*   __restrict__ dur,   // (B, TMAX)
                        const int*   __restrict__ plen,  // (B,)
                        float*       __restrict__ out)   // (B, TOUT, D)
{
    __shared__ int s_csum[LR_TMAX];
#if LR_ASYNC
    __shared__ float s_stage[LR_ROWS_PER_BLOCK][LR_D];   // 12 KB staging
#endif

    const int b  = blockIdx.y;
    const int i  = threadIdx.x;          // 0..255 == TMAX, one phoneme each
    const int pl = plen[b];

    // --- masked durations + inclusive scan (Hillis-Steele in LDS) ----------
    int d = (i < pl) ? dur[b * LR_TMAX + i] : 0;
    s_csum[i] = d;
    __syncthreads();
#pragma unroll
    for (int off = 1; off < LR_TMAX; off <<= 1) {
        int v   = s_csum[i];
        int add = (i >= off) ? s_csum[i - off] : 0;
        __syncthreads();
        s_csum[i] = v + add;
        __syncthreads();
    }
    int total = s_csum[LR_TMAX - 1];
    __syncthreads();                     // all reads done before fallback writes
    if (total == 0) {                    // reference: all-zero durations -> valid=1
        s_csum[i] = (i < pl) ? (i + 1) : pl;   // min(i+1, plen)
        total = pl;
    }
    __syncthreads();

    // --- one output frame per wave32 ---------------------------------------
    const int wave = threadIdx.x >> 5;
    const int lane = threadIdx.x & 31;
    const int t    = blockIdx.x * LR_ROWS_PER_BLOCK + wave;   // 0..TOUT-1

    // idx = upper_bound(csum, t) == (csum <= t).sum()  (uniform across wave;
    // same-address LDS reads broadcast conflict-free)
    int lo = 0, hi = LR_TMAX;
    while (lo < hi) {
        int mid = (lo + hi) >> 1;
        if (s_csum[mid] <= t) lo = mid + 1; else hi = mid;
    }
    const int idx = (lo < LR_TMAX - 1) ? lo : (LR_TMAX - 1);

    float4* dst = (float4*)(out + ((size_t)b * LR_TOUT + t) * LR_D);

    if (t < total) {                     // wave-uniform: EXEC all-ones here
        const float4* src =
            (const float4*)(emb + ((size_t)b * LR_TMAX + idx) * LR_D);
#if LR_ASYNC
        // 128-bit async gather: global -> LDS -> global, tracked by ASYNCcnt.
        // 3 x b128 per lane = 1536B row; 512B per instruction per wave.
        LR_AS3 float* stage3 = (LR_AS3 float*)&s_stage[wave][0];
#pragma unroll
        for (int k = 0; k < LR_F4_PER_LANE; ++k) {
            int e = lane + 32 * k;
            __builtin_amdgcn_global_load_async_to_lds_b128(
                (LR_AS1 lr_v4i*)(uintptr_t)(src + e),
                (LR_AS3 lr_v4i*)(stage3 + e * 4),
                0, 0);
        }
        __builtin_amdgcn_s_wait_asynccnt(0);  // LDS writes visible before reads
#pragma unroll
        for (int k = 0; k < LR_F4_PER_LANE; ++k) {
            int e = lane + 32 * k;
            __builtin_amdgcn_global_store_async_from_lds_b128(
                (LR_AS1 lr_v4i*)(uintptr_t)(dst + e),
                (LR_AS3 lr_v4i*)(stage3 + e * 4),
                0, 0);
        }
        // s_endpgm performs an implicit wait-idle; ASYNCcnt drains before exit.
#else
#pragma unroll
        for (int k = 0; k < LR_F4_PER_LANE; ++k) {
            int e = lane + 32 * k;
            dst[e] = src[e];
        }
#endif
    } else {                             // padded frames: write PAD_VALUE=0
        float4 z = make_float4(0.f, 0.f, 0.f, 0.f);
#pragma unroll
        for (int k = 0; k < LR_F4_PER_LANE; ++k) {
            dst[lane + 32 * k] = z;
        }
    }
}

extern "C" void kernel_launch(void* const* d_in, const int* in_sizes, int n_in,
                              void* d_out, int out_size, void* d_ws, size_t ws_size,
                              hipStream_t stream)
{
    (void)in_sizes; (void)n_in; (void)out_size; (void)d_ws; (void)ws_size;

    const float* emb  = (const float*)d_in[0];  // pho_embd_batch  (B,TMAX,D) f32
    const int*   dur  = (const int*)  d_in[1];  // durations_batch (B,TMAX)   i32
    const int*   plen = (const int*)  d_in[2];  // pho_len_batch   (B,)       i32
    float*       out  = (float*)      d_out;    // (B,TOUT,D) f32

    dim3 grid(LR_TOUT / LR_ROWS_PER_BLOCK, LR_B);  // (256, 32)
    dim3 block(256);                                // 8 wave32s
    length_regulator_kernel<<<grid, block, 0, stream>>>(emb, dur, plen, out);
}